// GlobalLocalAttention_49185965473823
// MI455X (gfx1250) — compile-verified
//
#include <hip/hip_runtime.h>
#include <hip/hip_bf16.h>

// GlobalLocalAttention for MI455X (gfx1250), fp32 WMMA (V_WMMA_F32_16X16X4_F32).
// B=4, N=1024, D=192. All GEMM-shaped work runs on the matrix pipe.

typedef float v2f __attribute__((ext_vector_type(2)));
typedef float v8f __attribute__((ext_vector_type(8)));

#define B_ 4
#define N_ 1024
#define D_ 192
#define QKVLD 576          // 3*D, row stride of qkv workspace
#define SCALE_ 0.07216878364870323f   // 192^-0.5

__device__ __forceinline__ v8f wmma4(v2f a, v2f b, v8f c) {
  // D = A(16x4) * B(4x16) + C(16x16), fp32 exact
  return __builtin_amdgcn_wmma_f32_16x16x4_f32(
      /*neg_a=*/false, a, /*neg_b=*/false, b,
      /*c_mod=*/(short)0, c, /*reuse_a=*/false, /*reuse_b=*/false);
}

__device__ __forceinline__ v8f vz8() { v8f v = {0.f,0.f,0.f,0.f,0.f,0.f,0.f,0.f}; return v; }

__device__ __forceinline__ int refl(int i) {   // jnp.pad mode='reflect'
  i = (i < 0) ? -i : i;
  i = (i >= N_) ? (2 * N_ - 2 - i) : i;
  return i;
}

// reductions within each 16-lane half (rows live per-half in C/D layout)
__device__ __forceinline__ float rmax16(float v) {
  v = fmaxf(v, __shfl_xor(v, 1, 32));
  v = fmaxf(v, __shfl_xor(v, 2, 32));
  v = fmaxf(v, __shfl_xor(v, 4, 32));
  v = fmaxf(v, __shfl_xor(v, 8, 32));
  return v;
}
__device__ __forceinline__ float rsum16(float v) {
  v += __shfl_xor(v, 1, 32);
  v += __shfl_xor(v, 2, 32);
  v += __shfl_xor(v, 4, 32);
  v += __shfl_xor(v, 8, 32);
  return v;
}

// ---------------- K1: qkv = x(4096x192) @ w_qkv(192x576) ----------------
__global__ __launch_bounds__(32) void k_qkv(const float* __restrict__ x,
                                            const float* __restrict__ w,
                                            float* __restrict__ qkv) {
  const int lane = threadIdx.x & 31;
  const int l16  = lane & 15;
  const int grp  = lane >> 4;
  const int tn   = blockIdx.x % 36;   // 576/16 col tiles
  const int tm   = blockIdx.x / 36;   // 4096/16 row tiles
  const int col  = tn * 16 + l16;
  const int arow = tm * 16 + l16;

  v8f acc = vz8();
#pragma unroll
  for (int kk = 0; kk < D_; kk += 4) {
    const int ka = kk + 2 * grp;
    v2f a = *(const v2f*)(x + arow * D_ + ka);          // A: row in lane
    v2f b;
    b.x = w[(ka + 0) * QKVLD + col];                    // B: K rows split by half-wave
    b.y = w[(ka + 1) * QKVLD + col];
    acc = wmma4(a, b, acc);
  }
#pragma unroll
  for (int r = 0; r < 8; ++r)
    qkv[(tm * 16 + r + 8 * grp) * QKVLD + col] = acc[r];
}

// ---------------- K2: local windowed attention (band GEMM trick) ----------------
// scores(16x192) = Qe(16x208) @ Kp[n0:n0+208](208x192), Qe[m,t]=q[m,t-m] in-band
// O_e(16x208)    = attn(16x192) @ Vp^T ; out_local[m,w] = O_e[m, m+w]
__global__ __launch_bounds__(32) void k_local(const float* __restrict__ qkv,
                                              float* __restrict__ out_local) {
  __shared__ __align__(16) float attn_lds[16 * D_];   // 12 KB
  const int lane = threadIdx.x & 31;
  const int l16  = lane & 15;
  const int grp  = lane >> 4;
  const int b    = blockIdx.x >> 6;
  const int n0   = (blockIdx.x & 63) * 16;

  const float* qb = qkv + (size_t)b * N_ * QKVLD;       // q at +0
  const float* kb = qb + D_;                            // k at +192
  const float* vb = qb + 2 * D_;                        // v at +384

  // ---- scores GEMM: contraction over t (208), 12 d-col blocks ----
  v8f acc[12];
#pragma unroll
  for (int j = 0; j < 12; ++j) acc[j] = vz8();

  for (int kk = 0; kk < 208; kk += 4) {
    const int t0 = kk + 2 * grp;
    const int m  = l16;
    const int d0 = t0 - m;
    v2f a;
    a.x = (d0 >= 0 && d0 < D_)     ? qb[(n0 + m) * QKVLD + d0]     : 0.f;
    a.y = (d0 + 1 >= 0 && d0 + 1 < D_) ? qb[(n0 + m) * QKVLD + d0 + 1] : 0.f;
    const int r0 = refl(n0 + t0 - 96);
    const int r1 = refl(n0 + t0 + 1 - 96);
#pragma unroll
    for (int j = 0; j < 12; ++j) {
      v2f bv;
      bv.x = kb[r0 * QKVLD + j * 16 + l16];
      bv.y = kb[r1 * QKVLD + j * 16 + l16];
      acc[j] = wmma4(a, bv, acc[j]);
    }
  }

  // ---- softmax over d (cols = lanes x 12 blocks) ----
  float inv[8];
#pragma unroll
  for (int c = 0; c < 8; ++c) {
    float mx = -3.0e38f;
#pragma unroll
    for (int j = 0; j < 12; ++j) mx = fmaxf(mx, acc[j][c] * SCALE_);
    mx = rmax16(mx);
    float s = 0.f;
#pragma unroll
    for (int j = 0; j < 12; ++j) {
      float e = __expf(acc[j][c] * SCALE_ - mx);
      acc[j][c] = e;
      s += e;
    }
    inv[c] = 1.f / rsum16(s);
  }
  // D-layout -> LDS (row-major attn tile) for A-operand reload
#pragma unroll
  for (int j = 0; j < 12; ++j)
#pragma unroll
    for (int c = 0; c < 8; ++c)
      attn_lds[(c + 8 * grp) * D_ + j * 16 + l16] = acc[j][c] * inv[c];
  __syncthreads();

  // ---- O_e GEMM: 13 t-blocks, contraction over d (192) ----
  for (int j = 0; j < 13; ++j) {
    const int vrow = refl(n0 + j * 16 + l16 - 96);
    v8f o = vz8();
    for (int kk = 0; kk < D_; kk += 4) {
      const int d = kk + 2 * grp;
      v2f a  = *(const v2f*)&attn_lds[l16 * D_ + d];
      v2f bv = *(const v2f*)&vb[vrow * QKVLD + d];
      o = wmma4(a, bv, o);
    }
#pragma unroll
    for (int c = 0; c < 8; ++c) {
      const int row = c + 8 * grp;
      const int w   = j * 16 + l16 - row;
      if ((unsigned)w < (unsigned)D_)
        out_local[((size_t)b * N_ + n0 + row) * D_ + w] = o[c];
    }
  }
}

// ---------------- K3: global attention, flash-style per 16-row q tile ----------------
__global__ __launch_bounds__(32) void k_global(const float* __restrict__ qkv,
                                               float* __restrict__ out_global) {
  __shared__ __align__(16) float p_lds[16 * 16];
  const int lane = threadIdx.x & 31;
  const int l16  = lane & 15;
  const int grp  = lane >> 4;
  const int b    = blockIdx.x >> 6;
  const int n0   = (blockIdx.x & 63) * 16;

  const float* qb = qkv + (size_t)b * N_ * QKVLD;
  const float* kb = qb + D_;
  const float* vb = qb + 2 * D_;

  v8f oacc[12];
#pragma unroll
  for (int j = 0; j < 12; ++j) oacc[j] = vz8();
  float m[8], l[8];
#pragma unroll
  for (int c = 0; c < 8; ++c) { m[c] = -3.0e38f; l[c] = 0.f; }

  for (int jk = 0; jk < 64; ++jk) {
    // S tile = Q(16x192) @ K^T
    v8f s = vz8();
    for (int kk = 0; kk < D_; kk += 4) {
      const int d = kk + 2 * grp;
      v2f a  = *(const v2f*)&qb[(n0 + l16) * QKVLD + d];
      v2f bv = *(const v2f*)&kb[(jk * 16 + l16) * QKVLD + d];
      s = wmma4(a, bv, s);
    }
    __syncthreads();   // WAR: previous iteration's P reads done before rewrite
    float alpha[8];
#pragma unroll
    for (int c = 0; c < 8; ++c) {
      float ms   = s[c] * SCALE_;
      float mnew = fmaxf(m[c], rmax16(ms));
      float p    = __expf(ms - mnew);
      alpha[c]   = __expf(m[c] - mnew);
      l[c]       = l[c] * alpha[c] + rsum16(p);
      m[c]       = mnew;
      p_lds[(c + 8 * grp) * 16 + l16] = p;   // D-layout -> row-major P
    }
#pragma unroll
    for (int j = 0; j < 12; ++j)
#pragma unroll
      for (int c = 0; c < 8; ++c) oacc[j][c] *= alpha[c];
    __syncthreads();   // RAW: P visible before A-operand reads
    // O += P(16x16) @ V(16x192)
#pragma unroll
    for (int kk = 0; kk < 16; kk += 4) {
      v2f a = *(const v2f*)&p_lds[l16 * 16 + kk + 2 * grp];
      const int r0 = jk * 16 + kk + 2 * grp;
#pragma unroll
      for (int j = 0; j < 12; ++j) {
        v2f bv;
        bv.x = vb[(r0 + 0) * QKVLD + j * 16 + l16];
        bv.y = vb[(r0 + 1) * QKVLD + j * 16 + l16];
        oacc[j] = wmma4(a, bv, oacc[j]);
      }
    }
  }
  float linv[8];
#pragma unroll
  for (int c = 0; c < 8; ++c) linv[c] = 1.f / l[c];
#pragma unroll
  for (int j = 0; j < 12; ++j)
#pragma unroll
    for (int c = 0; c < 8; ++c)
      out_global[((size_t)b * N_ + n0 + c + 8 * grp) * D_ + j * 16 + l16] =
          oacc[j][c] * linv[c];
}

// ---------------- K4: gate MLP + blend + output projection ----------------
__global__ __launch_bounds__(32) void k_gate_proj(const float* __restrict__ out_local,
                                                  const float* __restrict__ out_global,
                                                  const float* __restrict__ w_g1,
                                                  const float* __restrict__ b_g1,
                                                  const float* __restrict__ w_g2,
                                                  const float* __restrict__ b_g2,
                                                  const float* __restrict__ w_proj,
                                                  const float* __restrict__ b_proj,
                                                  float* __restrict__ out) {
  __shared__ __align__(16) float gate_lds[16];
  const int lane = threadIdx.x & 31;
  const int l16  = lane & 15;
  const int grp  = lane >> 4;
  const int b    = blockIdx.x >> 6;
  const int n0   = (blockIdx.x & 63) * 16;
  const size_t rb = (size_t)b * N_ + n0;

  // h = relu(concat(local,global) @ w_g1 + b_g1); fold into gsc = h @ w_g2
  float gpart[8];
#pragma unroll
  for (int c = 0; c < 8; ++c) gpart[c] = 0.f;

  for (int j = 0; j < 12; ++j) {
    v8f h = vz8();
    for (int kk = 0; kk < 2 * D_; kk += 4) {
      const int t = kk + 2 * grp;                 // even; pair never straddles 192
      const float* src = (t < D_) ? out_local : out_global;
      const int tt = (t < D_) ? t : (t - D_);
      v2f a = *(const v2f*)&src[(rb + l16) * D_ + tt];
      v2f bv;
      bv.x = w_g1[(t + 0) * D_ + j * 16 + l16];
      bv.y = w_g1[(t + 1) * D_ + j * 16 + l16];
      h = wmma4(a, bv, h);
    }
    const float wg2 = w_g2[j * 16 + l16];
    const float bg1 = b_g1[j * 16 + l16];
#pragma unroll
    for (int c = 0; c < 8; ++c)
      gpart[c] += fmaxf(h[c] + bg1, 0.f) * wg2;
  }
  const float bg2 = b_g2[0];
#pragma unroll
  for (int c = 0; c < 8; ++c) {
    float g = rsum16(gpart[c]) + bg2;
    g = 1.f / (1.f + __expf(-g));
    if (l16 == 0) gate_lds[c + 8 * grp] = g;      // lane0 -> rows 0..7, lane16 -> rows 8..15
  }
  __syncthreads();
  const float gm = gate_lds[l16];                 // gate for this lane's A-row

  // out = (g*local + (1-g)*global) @ w_proj + b_proj   (24 col blocks of 16)
  for (int j = 0; j < 24; ++j) {
    v8f o = vz8();
    for (int kk = 0; kk < D_; kk += 4) {
      const int d = kk + 2 * grp;
      v2f al = *(const v2f*)&out_local [(rb + l16) * D_ + d];
      v2f ag = *(const v2f*)&out_global[(rb + l16) * D_ + d];
      v2f a;
      a.x = gm * al.x + (1.f - gm) * ag.x;
      a.y = gm * al.y + (1.f - gm) * ag.y;
      v2f bv;
      bv.x = w_proj[(d + 0) * (2 * D_) + j * 16 + l16];
      bv.y = w_proj[(d + 1) * (2 * D_) + j * 16 + l16];
      o = wmma4(a, bv, o);
    }
    const float bp = b_proj[j * 16 + l16];
#pragma unroll
    for (int c = 0; c < 8; ++c)
      out[(rb + c + 8 * grp) * (2 * D_) + j * 16 + l16] = o[c] + bp;
  }
}

extern "C" void kernel_launch(void* const* d_in, const int* in_sizes, int n_in,
                              void* d_out, int out_size, void* d_ws, size_t ws_size,
                              hipStream_t stream) {
  (void)in_sizes; (void)n_in; (void)out_size; (void)ws_size;
  const float* x      = (const float*)d_in[0];
  const float* w_qkv  = (const float*)d_in[1];
  const float* w_proj = (const float*)d_in[2];
  const float* b_proj = (const float*)d_in[3];
  const float* w_g1   = (const float*)d_in[4];
  const float* b_g1   = (const float*)d_in[5];
  const float* w_g2   = (const float*)d_in[6];
  const float* b_g2   = (const float*)d_in[7];

  float* ws         = (float*)d_ws;
  float* qkv        = ws;                                    // 4096*576 f32 (9.4 MB)
  float* out_local  = qkv + (size_t)B_ * N_ * QKVLD;         // 4096*192
  float* out_global = out_local + (size_t)B_ * N_ * D_;      // 4096*192

  // One wave (32 threads) per 16-row tile; EXEC all-ones around every WMMA.
  k_qkv     <<<(B_ * N_ / 16) * (QKVLD / 16), 32, 0, stream>>>(x, w_qkv, qkv);
  k_local   <<<B_ * N_ / 16, 32, 0, stream>>>(qkv, out_local);
  k_global  <<<B_ * N_ / 16, 32, 0, stream>>>(qkv, out_global);
  k_gate_proj<<<B_ * N_ / 16, 32, 0, stream>>>(out_local, out_global,
                                               w_g1, b_g1, w_g2, b_g2,
                                               w_proj, b_proj, (float*)d_out);
}